// GraphDownSample_12120397709981
// MI455X (gfx1250) — compile-verified
//
#include <hip/hip_runtime.h>
#include <stdint.h>

// y[e,o,g] = sum_{c,k} x[e, c, 2g+k] * W[g, o, 2c+k] + b[g,o]
// e in [0, 64*16384); x record = 66 f32, y record = 33 f32.
//
// Roofline: ~415 MB moved, ~0.42 GFLOP -> ~1 flop/byte: HBM-bound, ~18us floor
// at 23.3 TB/s. Plan:
//   * double-buffered GLOBAL_LOAD_ASYNC_TO_LDS_B128 streaming (ASYNCcnt);
//     last iteration skips the prefetch (uniform branch) and waits asynccnt 0
//   * V_WMMA_F32_16X16X4_F32 per group: K=0..3 then K=4..5, with a ones
//     column at K=6 whose B-row holds the bias (bias fused into the GEMM)
//   * A-fragments come straight off ds_load_b64 (no permutes); loads are
//     software-pipelined one group ahead to hide LDS load-to-use latency
//   * D cols 0..2 staged in LDS, then coalesced b128 stores (tile base is
//     2112*t bytes = 16B aligned, so b128 is legal)

typedef __attribute__((ext_vector_type(2))) float v2f;
typedef __attribute__((ext_vector_type(4))) float v4f;
typedef __attribute__((ext_vector_type(8))) float v8f;

#define WAVES_PER_BLOCK 4
#define BDIM (WAVES_PER_BLOCK * 32)
#define TILES_PER_WAVE 8
#define NBLOCKS 2048
// per-wave LDS (dwords): two 16x66 input buffers + one 16x33 output stage
#define IN_BUF_DW   1056
#define OUT_OFF_DW  2112
#define WAVE_LDS_DW 2640

__device__ __forceinline__ void async_cp16(const void* gsrc, uint32_t lds_byte) {
  asm volatile("global_load_async_to_lds_b128 %0, %1, off"
               :: "v"(lds_byte), "v"(gsrc)
               : "memory");
}

__global__ __launch_bounds__(BDIM) void graph_downsample_wmma(
    const float* __restrict__ x, const float* __restrict__ W,
    const float* __restrict__ bias, float* __restrict__ y)
{
  __shared__ __attribute__((aligned(16))) float smem[WAVES_PER_BLOCK * WAVE_LDS_DW];
  const int lane = threadIdx.x & 31;
  const int wave = threadIdx.x >> 5;
  const int col  = lane & 15;   // N-index of B/C/D, record index of A
  const int half = lane >> 4;

  float* wmem = smem + wave * WAVE_LDS_DW;
  const uint32_t wbyte = (uint32_t)(uintptr_t)wmem; // low 32 bits == LDS byte addr

  // B fragments (4x16 f32): vgpr0 = rows {K0 | K2}, vgpr1 = rows {K1 | K3}.
  // Second wmma rows K'=0..3 map to i=4, i=5, ones(bias), zero.
  v2f B1[11], B2[11];
#pragma unroll
  for (int g = 0; g < 11; ++g) {
    float w0=0.f,w1=0.f,w2=0.f,w3=0.f,w4=0.f,w5=0.f,bb=0.f;
    if (col < 3) {
      const float* wp = W + g*18 + col*6;
      w0=wp[0]; w1=wp[1]; w2=wp[2]; w3=wp[3]; w4=wp[4]; w5=wp[5];
      bb = bias[g*3 + col];
    }
    B1[g].x = half ? w2  : w0;
    B1[g].y = half ? w3  : w1;
    B2[g].x = half ? bb  : w4;   // K'=2 row = bias (A column is all-ones)
    B2[g].y = half ? 0.f : w5;   // K'=3 row = 0
  }

  // A fragment source offsets inside a 16x66-record block (dwords; +2g at use)
  const int a1_off = col*66 + (half ? 22 : 0); // (i0,i1) | (i2,i3)
  const int a2_off = col*66 + 44;              // (i4,i5) | (dummy x 0-weight)

  const int gwave = blockIdx.x * WAVES_PER_BLOCK + wave;
  const long e0 = (long)gwave * (TILES_PER_WAVE * 16);

  // prefetch tile 0 into buffer 0 (1056 dwords = 264 x b128 = 8*32 + 8 lanes)
  {
    const char* src = (const char*)(x + e0*66);
#pragma unroll
    for (int i = 0; i < 8; ++i)
      async_cp16(src + (i*32 + lane)*16, wbyte + (uint32_t)((i*32 + lane)*16));
    if (lane < 8)
      async_cp16(src + (256 + lane)*16, wbyte + (uint32_t)((256 + lane)*16));
  }

  for (int it = 0; it < TILES_PER_WAVE; ++it) {
    const int buf  = it & 1;
    const int nbuf = buf ^ 1;

    if (it + 1 < TILES_PER_WAVE) {
      // prefetch next tile into the other buffer
      const char* src = (const char*)(x + (e0 + (long)(it + 1) * 16) * 66);
      const uint32_t ldsb = wbyte + (uint32_t)(nbuf * IN_BUF_DW * 4);
#pragma unroll
      for (int i = 0; i < 8; ++i)
        async_cp16(src + (i*32 + lane)*16, ldsb + (uint32_t)((i*32 + lane)*16));
      if (lane < 8)
        async_cp16(src + (256 + lane)*16, ldsb + (uint32_t)((256 + lane)*16));
      // newest 9 (next tile) may remain outstanding; current buffer is done
      asm volatile("s_wait_asynccnt 9" ::: "memory");
    } else {
      // nothing in flight except the current buffer's 9
      asm volatile("s_wait_asynccnt 0" ::: "memory");
    }

    const float* in = wmem + buf * IN_BUF_DW;
    float* outb     = wmem + OUT_OFF_DW;

    // Software-pipelined A-fragment loads: group g+1 is in flight while
    // group g computes (keeps ds_load -> v_wmma distance ~one group).
    v2f a1 = *(const v2f*)(in + a1_off);
    v2f a2 = *(const v2f*)(in + a2_off);
#pragma unroll
    for (int g = 0; g < 11; ++g) {
      v2f na1 = a1, na2 = a2;
      if (g < 10) {
        na1 = *(const v2f*)(in + a1_off + 2*(g + 1));
        na2 = *(const v2f*)(in + a2_off + 2*(g + 1));
      }
      if (half) a2.x = 1.0f;               // ones column K'=2 -> adds bias
      v8f acc = {};
      acc = __builtin_amdgcn_wmma_f32_16x16x4_f32(false, a1, false, B1[g],
                                                  (short)0, acc, false, false);
      acc = __builtin_amdgcn_wmma_f32_16x16x4_f32(false, a2, false, B2[g],
                                                  (short)0, acc, false, false);
      if (col < 3) {
        // D layout: vgpr r, lanes<16: (M=r, N=lane); lanes>=16: (M=r+8, N=lane-16)
        float* op = outb + col*11 + g + (half ? 8*33 : 0);
#pragma unroll
        for (int r = 0; r < 8; ++r) op[r*33] = acc[r];
      }
      a1 = na1;
      a2 = na2;
    }

    // coalesced write-out: 528 dwords = 132 x b128 = 4*32 + 4 lanes
    // (DS ops are in-order within a wave; outb is private to this wave)
    float* yp = y + (e0 + (long)it*16) * 33;   // byte base 2112*t: 16B aligned
#pragma unroll
    for (int i = 0; i < 4; ++i) {
      const int j = i*32 + lane;
      *(v4f*)(yp + 4*j) = *(const v4f*)(outb + 4*j);
    }
    if (lane < 4) {
      const int j = 128 + lane;
      *(v4f*)(yp + 4*j) = *(const v4f*)(outb + 4*j);
    }
  }
}

extern "C" void kernel_launch(void* const* d_in, const int* in_sizes, int n_in,
                              void* d_out, int out_size, void* d_ws, size_t ws_size,
                              hipStream_t stream) {
  (void)in_sizes; (void)n_in; (void)d_ws; (void)ws_size; (void)out_size;
  const float* x = (const float*)d_in[0];
  const float* W = (const float*)d_in[1];
  const float* b = (const float*)d_in[2];
  // d_in[3] = groups == arange(22).reshape(11,2); mapping n = 2g+k is hardcoded.
  float* y = (float*)d_out;
  graph_downsample_wmma<<<NBLOCKS, BDIM, 0, stream>>>(x, W, b, y);
}